// KmeansModel_14078902796984
// MI455X (gfx1250) — compile-verified
//
#include <hip/hip_runtime.h>

// ---- problem constants --------------------------------------------------
constexpr int Nn   = 131072;   // rows of x
constexpr int Dd   = 768;      // feature dim
constexpr int Kk   = 2000;     // centroids
constexpr int Kpad = 2048;     // padded K (multiple of 512)
#define LDSTRIDE 772           // padded LDS row stride in floats

typedef float v2f __attribute__((ext_vector_type(2)));
typedef float v4f __attribute__((ext_vector_type(4)));
typedef float v8f __attribute__((ext_vector_type(8)));

// fp32 WMMA: D(16x16,f32) = A(16x4,f32) * B(4x16,f32) + C
__device__ __forceinline__ v8f wmma_f32(v2f a, v2f b, v8f c) {
  return __builtin_amdgcn_wmma_f32_16x16x4_f32(
      /*neg_a=*/false, a, /*neg_b=*/false, b,
      /*c_mod=*/(short)0, c, /*reuse_a=*/false, /*reuse_b=*/false);
}

// ---- kernel 1: cn[k] = 0.5*||c_k||^2, padded with +inf -------------------
__global__ void cnorm_half_kernel(const float* __restrict__ cent,
                                  float* __restrict__ cn) {
  int k = blockIdx.x * blockDim.x + threadIdx.x;
  if (k >= Kpad) return;
  if (k >= Kk) { cn[k] = __builtin_inff(); return; }
  float s = 0.f;
  for (int d = 0; d < Dd; ++d) {
    float c = cent[(size_t)d * Kk + k];   // coalesced: consecutive k per lane
    s = fmaf(c, c, s);
  }
  cn[k] = 0.5f * s;
}

// ---- kernel 2: fused GEMM + argmax --------------------------------------
// block = 256 threads (8 waves). Block tile: 64 rows x (all K).
// wave tile: 64(M) x 64(N) per pass; 8 waves -> 512 cols/pass, 4 passes.
__global__ void __launch_bounds__(256)
kmeans_assign_kernel(const float* __restrict__ x,
                     const float* __restrict__ cent,
                     const float* __restrict__ cnorm,
                     int* __restrict__ out) {
  extern __shared__ float smem[];
  float* ldsX = smem;                    // 64 x LDSTRIDE floats

  const int r0   = blockIdx.x * 64;
  const int tid  = threadIdx.x;
  const int lane = tid & 31;
  const int wave = tid >> 5;
  const int half = lane >> 4;            // 0: lanes 0-15, 1: lanes 16-31
  const int l16  = lane & 15;

  // ---- stage X tile (64 x 768 fp32) into LDS, read X from HBM exactly once
  for (int i = tid; i < 64 * (Dd / 4); i += 256) {
    int row = i / (Dd / 4);
    int c4  = i - row * (Dd / 4);
    v4f v = ((const v4f*)(x + (size_t)(r0 + row) * Dd))[c4];
    *(v4f*)(ldsX + row * LDSTRIDE + c4 * 4) = v;
  }
  __syncthreads();

  // A-fragment base pointers (ds_load_b64 per k-step: K pair {0,1}/{2,3})
  const float* aPtr[4];
#pragma unroll
  for (int mi = 0; mi < 4; ++mi)
    aPtr[mi] = ldsX + (mi * 16 + l16) * LDSTRIDE + 2 * half;

  float bestVal[4][8];
  int   bestIdx[4][8];
#pragma unroll
  for (int mi = 0; mi < 4; ++mi)
#pragma unroll
    for (int r = 0; r < 8; ++r) { bestVal[mi][r] = -__builtin_inff(); bestIdx[mi][r] = 0; }

  for (int p = 0; p < Kpad / 512; ++p) {
    const int colBase = p * 512 + wave * 64;

    // named pointers (NOT an array): keeps global addrspace inference ->
    // global_load_b32 instead of flat_load_b32 (flat also burns DScnt)
    const int c0 = colBase + l16;
    const int c1 = colBase + 16 + l16;
    const int c2 = colBase + 32 + l16;
    const int c3 = colBase + 48 + l16;
    const int c0c = c0 < Kk ? c0 : Kk - 1;   // clamp B loads (OOB cols)
    const int c1c = c1 < Kk ? c1 : Kk - 1;
    const int c2c = c2 < Kk ? c2 : Kk - 1;
    const int c3c = c3 < Kk ? c3 : Kk - 1;
    const float* bp0 = cent + (size_t)(2 * half) * Kk + c0c;
    const float* bp1 = cent + (size_t)(2 * half) * Kk + c1c;
    const float* bp2 = cent + (size_t)(2 * half) * Kk + c2c;
    const float* bp3 = cent + (size_t)(2 * half) * Kk + c3c;
    const float i0 = -cnorm[c0];             // +inf pad -> acc = -inf forever
    const float i1 = -cnorm[c1];
    const float i2 = -cnorm[c2];
    const float i3 = -cnorm[c3];

    // init accumulators to -0.5*||c||^2 (per-lane broadcast matches C layout)
    v8f acc[4][4];
#pragma unroll
    for (int mi = 0; mi < 4; ++mi)
#pragma unroll
      for (int r = 0; r < 8; ++r) {
        acc[mi][0][r] = i0;
        acc[mi][1][r] = i1;
        acc[mi][2][r] = i2;
        acc[mi][3][r] = i3;
      }

    // ---- D reduction: 192 k-steps of depth 4, 16 WMMAs each
#pragma unroll 4
    for (int d0 = 0; d0 < Dd; d0 += 4) {
      v2f b0, b1, b2, b3;
      b0.x = bp0[0]; b0.y = bp0[Kk];
      b1.x = bp1[0]; b1.y = bp1[Kk];
      b2.x = bp2[0]; b2.y = bp2[Kk];
      b3.x = bp3[0]; b3.y = bp3[Kk];
      bp0 += 4 * (size_t)Kk;
      bp1 += 4 * (size_t)Kk;
      bp2 += 4 * (size_t)Kk;
      bp3 += 4 * (size_t)Kk;
#pragma unroll
      for (int mi = 0; mi < 4; ++mi) {
        v2f a = *(const v2f*)(aPtr[mi] + d0);   // ds_load_b64
        acc[mi][0] = wmma_f32(a, b0, acc[mi][0]);
        acc[mi][1] = wmma_f32(a, b1, acc[mi][1]);
        acc[mi][2] = wmma_f32(a, b2, acc[mi][2]);
        acc[mi][3] = wmma_f32(a, b3, acc[mi][3]);
      }
    }

    // ---- fold this pass into the running per-row argmax
#pragma unroll
    for (int mi = 0; mi < 4; ++mi)
#pragma unroll
      for (int ni = 0; ni < 4; ++ni) {
        const int col = colBase + ni * 16 + l16;
#pragma unroll
        for (int r = 0; r < 8; ++r) {
          float v = acc[mi][ni][r];
          if (v > bestVal[mi][r]) { bestVal[mi][r] = v; bestIdx[mi][r] = col; }
        }
      }
  }

  __syncthreads();  // all waves done reading ldsX; reuse it for reduction

  float* redV = smem;
  int*   redI = (int*)(smem + 512);

  // cross-lane argmax within each 16-lane column group (tie -> lower index)
#pragma unroll
  for (int mi = 0; mi < 4; ++mi)
#pragma unroll
    for (int r = 0; r < 8; ++r) {
      float v = bestVal[mi][r];
      int   i = bestIdx[mi][r];
#pragma unroll
      for (int off = 8; off >= 1; off >>= 1) {
        float ov = __shfl_xor(v, off);
        int   oi = __shfl_xor(i, off);
        if (ov > v || (ov == v && oi < i)) { v = ov; i = oi; }
      }
      if (l16 == 0) {
        int row = mi * 16 + 8 * half + r;    // C layout: lanes 0-15 -> M=r, 16-31 -> M=r+8
        redV[wave * 64 + row] = v;
        redI[wave * 64 + row] = i;
      }
    }
  __syncthreads();

  // cross-wave argmax (waves cover disjoint column ranges), then store index
  if (tid < 64) {
    float bv = redV[tid];
    int   bi = redI[tid];
#pragma unroll
    for (int w = 1; w < 8; ++w) {
      float v = redV[w * 64 + tid];
      int   i = redI[w * 64 + tid];
      if (v > bv || (v == bv && i < bi)) { bv = v; bi = i; }
    }
    out[r0 + tid] = bi;
  }
}

// ---- host glue ----------------------------------------------------------
extern "C" void kernel_launch(void* const* d_in, const int* in_sizes, int n_in,
                              void* d_out, int out_size, void* d_ws, size_t ws_size,
                              hipStream_t stream) {
  (void)in_sizes; (void)n_in; (void)out_size; (void)ws_size;
  const float* x    = (const float*)d_in[0];   // [N, D] fp32
  const float* cent = (const float*)d_in[1];   // [D, K] fp32
  float* cnorm = (float*)d_ws;                 // Kpad floats of scratch
  int*   out   = (int*)d_out;                  // [N] int32 indices

  cnorm_half_kernel<<<Kpad / 256, 256, 0, stream>>>(cent, cnorm);

  size_t ldsBytes = (size_t)64 * LDSTRIDE * sizeof(float);  // ~193 KB of 320 KB/WGP
  kmeans_assign_kernel<<<Nn / 64, 256, ldsBytes, stream>>>(x, cent, cnorm, out);
}